// QModel_65481071407304
// MI455X (gfx1250) — compile-verified
//
#include <hip/hip_runtime.h>

typedef __attribute__((ext_vector_type(2))) float v2f;
typedef __attribute__((ext_vector_type(8))) float v8f;

#define N_WIRES  14
#define N_LAYERS 3
#define DIM      16384   // 2^14

// Combined gate U = RY(b) @ RX(a), entry (mb, nb):
//   U00 =  cb*ca + i*sb*sa     U01 = -sb*ca - i*cb*sa
//   U10 =  sb*ca - i*cb*sa     U11 =  cb*ca - i*sb*sa
__device__ __forceinline__ void uent(float ca, float sa, float cb, float sb,
                                     int mb, int nb, float& ur, float& ui) {
    if (mb == 0 && nb == 0)      { ur =  cb * ca; ui =  sb * sa; }
    else if (mb == 0)            { ur = -sb * ca; ui = -cb * sa; }
    else if (nb == 0)            { ur =  sb * ca; ui = -cb * sa; }
    else                         { ur =  cb * ca; ui = -sb * sa; }
}

// Precompute the fused 16x16 complex group matrices.
// Layout in workspace: G[((layer*4+group)*2 + re0im1)*256 + m*16 + n]
// group 0: wires 0-3  (state bits 13..10, wire0 = MSB of the 4-bit row index)
// group 1: wires 4-7  (bits 9..6)
// group 2: wires 8-11 (bits 5..2)
// group 3: I (x) I (x) U12 (x) U13 on bits 3..0 (identity part overlaps group 2 harmlessly)
__global__ void qgate_precompute(const float* __restrict__ params,
                                 float* __restrict__ G) {
    int tid = threadIdx.x;
    int m = tid >> 4, n = tid & 15;
    for (int l = 0; l < N_LAYERS; ++l) {
        for (int g = 0; g < 4; ++g) {
            float re = 1.0f, im = 0.0f;
            #pragma unroll
            for (int k = 3; k >= 0; --k) {          // k = bit within 4-bit group (3 = MSB)
                int mb = (m >> k) & 1, nb = (n >> k) & 1;
                float ur, ui;
                if (g == 3 && k >= 2) {             // identity factor
                    ur = (mb == nb) ? 1.0f : 0.0f; ui = 0.0f;
                } else {
                    int wire = (g < 3) ? (g * 4 + (3 - k)) : (12 + (1 - k));
                    float a = 0.5f * params[(l * N_WIRES + wire) * 2 + 0]; // RX angle/2
                    float b = 0.5f * params[(l * N_WIRES + wire) * 2 + 1]; // RY angle/2
                    float ca = cosf(a), sa = sinf(a);
                    float cb = cosf(b), sb = sinf(b);
                    uent(ca, sa, cb, sb, mb, nb, ur, ui);
                }
                float nre = re * ur - im * ui;
                float nim = re * ui + im * ur;
                re = nre; im = nim;
            }
            G[((l * 4 + g) * 2 + 0) * 256 + m * 16 + n] = re;
            G[((l * 4 + g) * 2 + 1) * 256 + m * 16 + n] = im;
        }
    }
}

// Composed source index for one layer of CNOT(0,1)..CNOT(12,13),CNOT(13,0).
// Wire w lives at bit (13-w). new[i] = old[src(i)].
__device__ __forceinline__ int cnot_src(int i) {
    i ^= (i & 1) << 13;                              // undo CNOT(13,0): pc=0, pt=13
    #pragma unroll
    for (int w = 12; w >= 0; --w)                    // undo CNOT(w,w+1) in reverse
        i ^= ((i >> (13 - w)) & 1) << (12 - w);
    return i;
}

__device__ __forceinline__ v8f wmma4(v2f a, v2f b, v8f c) {
    // (neg_a, A, neg_b, B, c_mod, C, reuse_a, reuse_b)
    return __builtin_amdgcn_wmma_f32_16x16x4_f32(false, a, false, b, (short)0, c,
                                                 false, false);
}

// One workgroup (8 wave32) per batch element. Dynamic LDS: two 128KB complex
// statevector buffers (re[16384] + im[16384] each) = 256 KB of the 320 KB WGP LDS.
__launch_bounds__(256)
__global__ void qsim_kernel(const float* __restrict__ state,
                            const float* __restrict__ G,
                            const float* __restrict__ head_w,
                            const float* __restrict__ head_b,
                            float* __restrict__ out) {
    extern __shared__ float lds[];
    const int b    = blockIdx.x;
    const int tid  = threadIdx.x;
    const int lane = tid & 31;
    const int wave = tid >> 5;

    // Load real statevector, im = 0. Buffer 0 at float-offset 0, buffer 1 at 2*DIM.
    for (int i = tid; i < DIM; i += 256) {
        lds[i]       = state[b * DIM + i];
        lds[DIM + i] = 0.0f;
    }
    __syncthreads();

    int cur = 0;
    for (int l = 0; l < N_LAYERS; ++l) {
        // ---- fused 4-wire gate groups via V_WMMA_F32_16X16X4_F32 ----
        for (int g = 0; g < 4; ++g) {
            float* re = lds + cur;
            float* im = lds + cur + DIM;
            const int sh = (g < 3) ? (10 - 4 * g) : 0;   // row stride = 1<<sh

            // A fragments (G matrix), shared by all tiles of this group/layer.
            // 16x4 f32 A: lane m (0-15): VGPR0=A[m,4q+0], VGPR1=A[m,4q+1];
            // lanes 16-31: K=4q+2, 4q+3.
            const float* Gre = G + ((l * 4 + g) * 2 + 0) * 256;
            const float* Gim = Gre + 256;
            const int am = lane & 15;
            const int kh = lane >> 4;
            v2f are[4], aim[4];
            #pragma unroll
            for (int q = 0; q < 4; ++q) {
                int k0 = 4 * q + 2 * kh;
                are[q].x = Gre[am * 16 + k0]; are[q].y = Gre[am * 16 + k0 + 1];
                aim[q].x = Gim[am * 16 + k0]; aim[q].y = Gim[am * 16 + k0 + 1];
            }

            // 64 column-tiles of 16, 8 per wave; tiles are independent -> in-place.
            for (int j = 0; j < 8; ++j) {
                const int t = wave * 8 + j;
                const int c = t * 16 + am;                       // column id 0..1023
                const int lomask = (1 << sh) - 1;
                const int base = ((c >> sh) << (sh + 4)) | (c & lomask);

                // B fragments: 4x16 f32 B: lane n: VGPR0=B[2*kh][n], VGPR1=B[2*kh+1][n]
                v2f bre[4], bim[4];
                #pragma unroll
                for (int q = 0; q < 4; ++q) {
                    int a0 = base + ((4 * q + 2 * kh) << sh);
                    int a1 = a0 + (1 << sh);
                    bre[q].x = re[a0]; bre[q].y = re[a1];
                    bim[q].x = im[a0]; bim[q].y = im[a1];
                }

                v8f P = {0.f,0.f,0.f,0.f,0.f,0.f,0.f,0.f};
                v8f Q = {0.f,0.f,0.f,0.f,0.f,0.f,0.f,0.f};
                v8f R = {0.f,0.f,0.f,0.f,0.f,0.f,0.f,0.f};
                #pragma unroll
                for (int q = 0; q < 4; ++q) P = wmma4(are[q], bre[q], P); // Gre*Xre
                #pragma unroll
                for (int q = 0; q < 4; ++q) Q = wmma4(aim[q], bim[q], Q); // Gim*Xim
                #pragma unroll
                for (int q = 0; q < 4; ++q) R = wmma4(are[q], bim[q], R); // Gre*Xim
                #pragma unroll
                for (int q = 0; q < 4; ++q) R = wmma4(aim[q], bre[q], R); // +Gim*Xre

                // D store per C/D layout: lane = N (+16 -> rows 8..15), VGPR v = row
                const int mb = kh * 8;
                #pragma unroll
                for (int v = 0; v < 8; ++v) {
                    int addr = base + ((mb + v) << sh);
                    re[addr] = P[v] - Q[v];   // real part
                    im[addr] = R[v];          // imag part
                }
            }
            __syncthreads();
        }

        // ---- CNOT ring: one composed permutation, gather cur -> other buffer ----
        {
            const int nxt = cur ^ (2 * DIM);
            const float* sre = lds + cur;
            const float* sim = lds + cur + DIM;
            float* dre = lds + nxt;
            float* dim_ = lds + nxt + DIM;
            for (int i = tid; i < DIM; i += 256) {
                int s = cnot_src(i);
                dre[i]  = sre[s];
                dim_[i] = sim[s];
            }
            cur = nxt;
            __syncthreads();
        }
    }

    // ---- <Z_w> features + linear head, fused: out = sum_i |psi_i|^2 * S(i) + b
    float hw[N_WIRES];
    #pragma unroll
    for (int w = 0; w < N_WIRES; ++w) hw[w] = head_w[w];

    const float* re = lds + cur;
    const float* im = lds + cur + DIM;
    float acc = 0.0f;
    for (int i = tid; i < DIM; i += 256) {
        float p = re[i] * re[i] + im[i] * im[i];
        float s = 0.0f;
        #pragma unroll
        for (int w = 0; w < N_WIRES; ++w)
            s += ((i >> (13 - w)) & 1) ? -hw[w] : hw[w];
        acc += p * s;
    }

    // Block reduction in the dead (non-cur) buffer.
    float* red = lds + (cur ^ (2 * DIM));
    red[tid] = acc;
    __syncthreads();
    #pragma unroll
    for (int sft = 128; sft > 0; sft >>= 1) {
        if (tid < sft) red[tid] += red[tid + sft];
        __syncthreads();
    }
    if (tid == 0) out[b] = red[0] + head_b[0];
}

extern "C" void kernel_launch(void* const* d_in, const int* in_sizes, int n_in,
                              void* d_out, int out_size, void* d_ws, size_t ws_size,
                              hipStream_t stream) {
    (void)n_in; (void)out_size; (void)ws_size;
    const float* state  = (const float*)d_in[0];   // (B, 16384) f32
    const float* params = (const float*)d_in[1];   // (3, 14, 2) f32
    const float* hw     = (const float*)d_in[2];   // (1, 14)    f32
    const float* hb     = (const float*)d_in[3];   // (1,)       f32
    float* out = (float*)d_out;                    // (B,)       f32
    float* G   = (float*)d_ws;                     // 6144 floats of fused gates

    const int batch = in_sizes[0] / DIM;

    qgate_precompute<<<1, 256, 0, stream>>>(params, G);
    qsim_kernel<<<batch, 256, 4 * DIM * sizeof(float), stream>>>(state, G, hw, hb, out);
}